// GraphSAGELayer_67920612819026
// MI455X (gfx1250) — compile-verified
//
#include <hip/hip_runtime.h>

// GraphSAGE fused kernel for gfx1250 (MI455X), wave32 + V_WMMA_F32_16X16X4_F32.
//
// combined[B,1024] = [ x[batch_nodes] | mean_s x[neigh_idx[:,s]] ]
// out = relu(combined @ W[1024,512])
//
// Block = 256 threads (8 wave32) -> 32-row x 512-col output tile.
// Phase 1: gather self + mean(neighbors) -> XOR-swizzled 32x1024 f32 A-tile in LDS.
// Phase 2: GEMM with f32 WMMA; W staged through LDS in double-buffered BK=8
//          chunks, k-pair interleaved + XOR-swizzled (conflict-free ds_load_b64).
//
// LDS = 131072 (A) + 32768 (B, 2 bufs) = 163840 B  ->  2 workgroups per WGP
// (2 x 163840 = 327680 = 320 KB), so one block's gather overlaps the
// co-resident block's WMMA phase.

typedef float v2f __attribute__((ext_vector_type(2)));
typedef float v8f __attribute__((ext_vector_type(8)));

#define IN_F     512
#define K_TOTAL  1024
#define OUT_F    512
#define BM       32
#define SAMPLE   10
#define BK       8                   // K chunk staged in LDS
#define B_PROW   1024                // dwords per k-pair row (512 cols x 2, exact)
#define B_BUF    ((BK / 2) * B_PROW) // 4096 dwords per chunk buffer
#define NCHUNK   (K_TOTAL / BK)      // 128

// A swizzle: element (r,k) at  r*1024 + (k ^ ((r&15)<<2))   [bank-conflict-free
// column reads, pairs (k,k+1) stay contiguous since mask has no bits 0..1]
// B swizzle: pair-element (p,n,k&1) at  p*B_PROW + ((2n) ^ ((p&1)<<5)) + (k&1)
// [adjacent pair-rows land on complementary 32-bank halves]

// Cooperative load of W[kk0..kk0+8][0..512) into LDS (k-pair interleaved).
__device__ __forceinline__
void load_w_chunk(const float* __restrict__ w, float* __restrict__ bsBuf,
                  int kk0, int tid)
{
    #pragma unroll
    for (int i = 0; i < 2; ++i) {
        const int task = i * 256 + tid;   // 512 tasks: 4 k-pairs x 128 col-quads
        const int kp = task >> 7;         // 0..3
        const int ng = task & 127;        // group of 4 columns
        const float4 w0 = *reinterpret_cast<const float4*>(
            w + (size_t)(kk0 + 2 * kp)     * OUT_F + 4 * ng);
        const float4 w1 = *reinterpret_cast<const float4*>(
            w + (size_t)(kk0 + 2 * kp + 1) * OUT_F + 4 * ng);
        float* dst = bsBuf + kp * B_PROW + ((8 * ng) ^ ((kp & 1) << 5));
        float4 p0; p0.x = w0.x; p0.y = w1.x; p0.z = w0.y; p0.w = w1.y;
        float4 p1; p1.x = w0.z; p1.y = w1.z; p1.z = w0.w; p1.w = w1.w;
        *reinterpret_cast<float4*>(dst)     = p0;   // ds_store_b128 (32B aligned)
        *reinterpret_cast<float4*>(dst + 4) = p1;
    }
}

__global__ __launch_bounds__(256)
void sage_fused_wmma_f32(const float* __restrict__ x,
                         const int*   __restrict__ neigh_idx,
                         const int*   __restrict__ batch_nodes,
                         const float* __restrict__ w,
                         float*       __restrict__ out,
                         int nB)
{
    __shared__ float As[BM * 1024];   // 131,072 B (swizzled, no padding)
    __shared__ float Bs[2 * B_BUF];   //  32,768 B (double buffer)

    const int tid       = threadIdx.x;
    const int blockRow0 = blockIdx.x * BM;

    // -------- Phase 1: gather self rows + mean(neighbor rows) into LDS ------
    {
        const int r    = tid >> 3;        // 0..31 : tile row
        const int l8   = tid & 7;         // 8 threads per row
        const int mask = (r & 15) << 2;   // A swizzle mask (bits 2..5)
        int gr = blockRow0 + r;
        if (gr >= nB) gr = nB - 1;        // clamp (stores are guarded below)

        const int     node = batch_nodes[gr];
        const float4* xrow = reinterpret_cast<const float4*>(x + (size_t)node * IN_F);
        float*        dstR = As + r * 1024;
        #pragma unroll 4
        for (int c = l8; c < IN_F / 4; c += 8) {
            float4 v = xrow[c];
            *reinterpret_cast<float4*>(dstR + ((4 * c) ^ mask)) = v;
        }

        const int* nidx = neigh_idx + (size_t)gr * SAMPLE;
        int idxs[SAMPLE];
        #pragma unroll
        for (int s = 0; s < SAMPLE; ++s) idxs[s] = nidx[s];

        const float invS = 1.0f / (float)SAMPLE;
        #pragma unroll 2
        for (int c = l8; c < IN_F / 4; c += 8) {
            float sx = 0.f, sy = 0.f, sz = 0.f, sw = 0.f;
            #pragma unroll
            for (int s = 0; s < SAMPLE; ++s) {
                const float4* nrow =
                    reinterpret_cast<const float4*>(x + (size_t)idxs[s] * IN_F);
                float4 v = nrow[c];
                sx += v.x; sy += v.y; sz += v.z; sw += v.w;
            }
            float4 m; m.x = sx * invS; m.y = sy * invS; m.z = sz * invS; m.w = sw * invS;
            *reinterpret_cast<float4*>(dstR + ((IN_F + 4 * c) ^ mask)) = m;
        }
    }

    // -------- Phase 2: C[32x512] = A[32x1024] x W[1024x512] via f32 WMMA ----
    const int wave  = tid >> 5;                  // 0..7
    const int lane  = tid & 31;
    const int waveM = wave >> 2;                 // 0..1 : 16-row half
    const int waveN = wave & 3;                  // 0..3 : 128-col strip
    const int rowA  = waveM * 16 + (lane & 15);  // A-matrix M = lane % 16
    const int koff  = (lane >> 4) << 1;          // lanes 0-15 -> K0/K1, 16-31 -> K2/K3
    const int nLane = waveN * 128 + (lane & 15); // B/C column (tile 0)
    const int amask = (lane & 15) << 2;          // per-lane A swizzle mask

    v8f acc[8];
    #pragma unroll
    for (int t = 0; t < 8; ++t)
        acc[t] = (v8f){0.f, 0.f, 0.f, 0.f, 0.f, 0.f, 0.f, 0.f};

    const float* aRow = As + rowA * 1024;

    // Preload first W chunk, and make phase-1 A-tile visible.
    load_w_chunk(w, Bs, 0, tid);
    __syncthreads();

    for (int c = 0; c < NCHUNK; ++c) {
        const int cur = c & 1;
        if (c + 1 < NCHUNK)
            load_w_chunk(w, Bs + (cur ^ 1) * B_BUF, (c + 1) * BK, tid);

        const float* bsCur = Bs + cur * B_BUF;
        const int    kk0   = c * BK;
        #pragma unroll
        for (int kk2 = 0; kk2 < BK; kk2 += 4) {
            // A fragment: {A[row][k], A[row][k+1]}, k = kk0+kk2+koff (even)
            const int k = kk0 + kk2 + koff;
            v2f a = *reinterpret_cast<const v2f*>(aRow + (k ^ amask));
            // B fragments from pair-row p of this chunk
            const int    p    = (kk2 + koff) >> 1;      // 0..3
            const int    m32  = (p & 1) << 5;
            const float* bp   = bsCur + p * B_PROW;
            #pragma unroll
            for (int t = 0; t < 8; ++t) {
                v2f b = *reinterpret_cast<const v2f*>(
                    bp + ((2 * (nLane + 16 * t)) ^ m32));   // ds_load_b64
                acc[t] = __builtin_amdgcn_wmma_f32_16x16x4_f32(
                    /*neg_a=*/false, a, /*neg_b=*/false, b,
                    /*c_mod=*/(short)0, acc[t],
                    /*reuse_a=*/false, /*reuse_b=*/false);
            }
        }
        __syncthreads();   // readers done with `cur`; writes to `cur^1` visible
    }

    // -------- Epilogue: ReLU + store (16x16 f32 C/D lane layout) ------------
    const int mOff = (lane >> 4) << 3;           // 0 or 8
    #pragma unroll
    for (int t = 0; t < 8; ++t) {
        const int n = nLane + t * 16;
        #pragma unroll
        for (int v = 0; v < 8; ++v) {
            const int m = blockRow0 + waveM * 16 + mOff + v;
            if (m < nB) {
                out[(size_t)m * OUT_F + n] = fmaxf(acc[t][v], 0.0f);
            }
        }
    }
}

extern "C" void kernel_launch(void* const* d_in, const int* in_sizes, int n_in,
                              void* d_out, int out_size, void* d_ws, size_t ws_size,
                              hipStream_t stream) {
    const float* x           = (const float*)d_in[0];   // [N_NODES, 512] f32
    const int*   neigh_idx   = (const int*)  d_in[1];   // [B, 10] int32
    const int*   batch_nodes = (const int*)  d_in[2];   // [B] int32
    const float* weight      = (const float*)d_in[3];   // [1024, 512] f32
    float*       out         = (float*)d_out;           // [B, 512] f32

    const int B    = in_sizes[2];                       // 50000
    const int grid = (B + BM - 1) / BM;                 // 1563

    sage_fused_wmma_f32<<<grid, 256, 0, stream>>>(x, neigh_idx, batch_nodes,
                                                  weight, out, B);
    (void)n_in; (void)out_size; (void)d_ws; (void)ws_size;
}